// GCNNodeClassifierNetwork_13383118094673
// MI455X (gfx1250) — compile-verified
//
#include <hip/hip_runtime.h>
#include <stdint.h>

typedef float v2f __attribute__((ext_vector_type(2)));
typedef float v8f __attribute__((ext_vector_type(8)));

#define NNODES 50000
#define NEDGES 1600000
#define DIN    128
#define DHID   256
#define DOUT   128

// ---------------------------------------------------------------------------
// Degree / normalization
// ---------------------------------------------------------------------------
__global__ void init_deg_kernel(float* __restrict__ deg, int n) {
  int i = blockIdx.x * blockDim.x + threadIdx.x;
  if (i < n) deg[i] = 1.0f;  // self loop contributes 1 to every node
}

__global__ void deg_accum_kernel(const int* __restrict__ dst,
                                 float* __restrict__ deg, int e) {
  int i = blockIdx.x * blockDim.x + threadIdx.x;
  if (i < e) atomicAdd(&deg[dst[i]], 1.0f);
}

__global__ void dinv_kernel(const float* __restrict__ deg,
                            float* __restrict__ dinv, int n) {
  int i = blockIdx.x * blockDim.x + threadIdx.x;
  if (i < n) {
    float d = deg[i];
    dinv[i] = (d > 0.0f) ? rsqrtf(d) : 0.0f;
  }
}

// ---------------------------------------------------------------------------
// fp32 WMMA GEMM with async-LDS A staging.
//   C[M x NC] = A[M x K] * B[K x NC]   (row-major, M%16==0)
//
// One block (128 threads = 4 wave32) owns a 16-row output panel:
//   - The 16xK A panel (contiguous 16*K floats) is staged into LDS once via
//     GLOBAL_LOAD_ASYNC_TO_LDS_B128 (ASYNCcnt), then s_wait_asynccnt + barrier.
//   - Each wave computes NT = NC/64 independent 16x16 tiles, sharing the LDS
//     A fragment across NT back-to-back V_WMMA_F32_16X16X4_F32 ops.
//
// 16x4 fp32 A layout (ISA 7.12.2): lanes 0-15 -> rows M, K=k0,k0+1;
// lanes 16-31 -> rows M, K=k0+2,k0+3.  B 4x16 mirrors it per lane-half.
// C/D: VGPR r -> (M=r, N=lane) lanes 0-15, (M=8+r, N=lane-16) lanes 16-31.
// EXEC all ones throughout (exact grid, uniform control flow) as WMMA needs.
// ---------------------------------------------------------------------------
template <int K, int NC>
__global__ void wmma_gemm_f32_lds_kernel(const float* __restrict__ A,
                                         const float* __restrict__ B,
                                         float* __restrict__ C) {
  constexpr int WAVES = 4;
  constexpr int NT = NC / (16 * WAVES);   // 16x16 tiles per wave
  __shared__ float Atile[16 * K];

  const int tid  = threadIdx.x;
  const int lane = tid & 31;
  const int wave = tid >> 5;
  const int m0   = blockIdx.x << 4;

  // ---- async stage of A panel: rows [m0, m0+16) are 16*K contiguous floats
  {
    const float* __restrict__ Abase = A + (long)m0 * K;
    constexpr int VEC4 = (16 * K) / 4;    // float4 chunks
#pragma unroll
    for (int c = tid; c < VEC4; c += 128) {
      uint32_t lds_off = (uint32_t)(uintptr_t)(&Atile[c << 2]);
      const float* g = Abase + (c << 2);
      asm volatile("global_load_async_to_lds_b128 %0, %1, off"
                   :: "v"(lds_off), "v"(g)
                   : "memory");
    }
    asm volatile("s_wait_asynccnt 0x0" ::: "memory");
    __syncthreads();
  }

  const bool hi = (lane >= 16);
  const int  l  = lane & 15;
  const int  ka = hi ? 2 : 0;
  const float* __restrict__ arow = &Atile[l * K];
  const int nbase = (wave * NT) << 4;     // first N column of this wave
  const int ncol0 = nbase + l;

  v8f acc[NT];
#pragma unroll
  for (int t = 0; t < NT; ++t)
    acc[t] = (v8f){0.f, 0.f, 0.f, 0.f, 0.f, 0.f, 0.f, 0.f};

  for (int k0 = 0; k0 < K; k0 += 4) {
    __builtin_prefetch(&B[(long)(k0 + 8) * NC + ncol0], 0, 1);  // global_prefetch_b8
    const v2f a = *(const v2f*)(arow + k0 + ka);                 // ds_load_b64
#pragma unroll
    for (int t = 0; t < NT; ++t) {
      const int ncol = ncol0 + (t << 4);
      v2f b;
      b.x = B[(long)(k0 + ka) * NC + ncol];
      b.y = B[(long)(k0 + ka + 1) * NC + ncol];
      acc[t] = __builtin_amdgcn_wmma_f32_16x16x4_f32(
          /*neg_a=*/false, a, /*neg_b=*/false, b,
          /*c_mod=*/(short)0, acc[t], /*reuse_a=*/false, /*reuse_b=*/false);
    }
  }

  float* __restrict__ crow = C + (long)(m0 + (hi ? 8 : 0)) * NC;
#pragma unroll
  for (int t = 0; t < NT; ++t) {
    const int ncol = ncol0 + (t << 4);
#pragma unroll
    for (int r = 0; r < 8; ++r) crow[(long)r * NC + ncol] = acc[t][r];
  }
}

// ---------------------------------------------------------------------------
// Aggregation (scatter-add) kernels
// ---------------------------------------------------------------------------

// AGG[i,f] = H[i,f] * dinv[i]^2   (self-loop message; also zero-initializes)
__global__ void self_init_kernel(const float* __restrict__ H,
                                 const float* __restrict__ dinv,
                                 float* __restrict__ AGG,
                                 int n, int logF) {
  long idx = (long)blockIdx.x * blockDim.x + threadIdx.x;
  long total = (long)n << logF;
  if (idx >= total) return;
  int i = (int)(idx >> logF);
  float di = dinv[i];
  AGG[idx] = H[idx] * di * di;
}

// AGG[dst,f] += H[src,f] * dinv[src]*dinv[dst]   for every (edge, feature)
__global__ void edge_agg_kernel(const int* __restrict__ src,
                                const int* __restrict__ dst,
                                const float* __restrict__ dinv,
                                const float* __restrict__ H,
                                float* __restrict__ AGG,
                                int e, int logF) {
  long idx = (long)blockIdx.x * blockDim.x + threadIdx.x;
  long total = (long)e << logF;
  if (idx >= total) return;
  int eid = (int)(idx >> logF);
  int f   = (int)(idx & ((1 << logF) - 1));
  int s = src[eid];
  int d = dst[eid];
  float nrm = dinv[s] * dinv[d];
  atomicAdd(&AGG[((long)d << logF) + f], H[((long)s << logF) + f] * nrm);
}

// In-place: AGG = relu(AGG + b[f])
__global__ void bias_relu_kernel(float* __restrict__ AGG,
                                 const float* __restrict__ b,
                                 int n, int logF) {
  long idx = (long)blockIdx.x * blockDim.x + threadIdx.x;
  long total = (long)n << logF;
  if (idx >= total) return;
  int f = (int)(idx & ((1 << logF) - 1));
  float v = AGG[idx] + b[f];
  AGG[idx] = v > 0.0f ? v : 0.0f;
}

// out[i,f] = H2[i,f]*dinv[i]^2 + b2[f] + x[i,f]   (self-loop + bias + residual)
__global__ void out_init_kernel(const float* __restrict__ H2,
                                const float* __restrict__ dinv,
                                const float* __restrict__ b2,
                                const float* __restrict__ x,
                                float* __restrict__ out,
                                int n, int logF) {
  long idx = (long)blockIdx.x * blockDim.x + threadIdx.x;
  long total = (long)n << logF;
  if (idx >= total) return;
  int i = (int)(idx >> logF);
  int f = (int)(idx & ((1 << logF) - 1));
  float di = dinv[i];
  out[idx] = H2[idx] * di * di + b2[f] + x[idx];
}

// ---------------------------------------------------------------------------
// Launcher
// ---------------------------------------------------------------------------
extern "C" void kernel_launch(void* const* d_in, const int* in_sizes, int n_in,
                              void* d_out, int out_size, void* d_ws, size_t ws_size,
                              hipStream_t stream) {
  const float* x  = (const float*)d_in[0];                 // [N, 128]
  const int*   ei = (const int*)d_in[1];                   // [2, E]
  const float* W1 = (const float*)d_in[2];                 // [128, 256]
  const float* b1 = (const float*)d_in[3];                 // [256]
  const float* W2 = (const float*)d_in[4];                 // [256, 128]
  const float* b2 = (const float*)d_in[5];                 // [128]
  float* out = (float*)d_out;                              // [N, 128]

  const int* src = ei;            // edge_index[0]
  const int* dst = ei + NEDGES;   // edge_index[1]

  // Workspace layout (floats):
  //   deg  : N
  //   dinv : N
  //   H    : N*256  (layer-1 transform; reused for layer-2 H2 which is N*128)
  //   AGG  : N*256  (layer-1 aggregation -> h1 after bias+relu)
  float* ws   = (float*)d_ws;
  float* deg  = ws;
  float* dinv = ws + NNODES;
  float* H    = ws + 2 * (long)NNODES;
  float* AGG  = H + (long)NNODES * DHID;

  const int TB = 256;

  // --- degree & symmetric normalization -----------------------------------
  init_deg_kernel<<<(NNODES + TB - 1) / TB, TB, 0, stream>>>(deg, NNODES);
  deg_accum_kernel<<<(NEDGES + TB - 1) / TB, TB, 0, stream>>>(dst, deg, NEDGES);
  dinv_kernel<<<(NNODES + TB - 1) / TB, TB, 0, stream>>>(deg, dinv, NNODES);

  // --- layer 1: H = X @ W1  (WMMA f32 + async-LDS staging), then aggregate
  wmma_gemm_f32_lds_kernel<DIN, DHID><<<NNODES / 16, 128, 0, stream>>>(x, W1, H);
  {
    long total = (long)NNODES << 8;  // N * 256
    self_init_kernel<<<(int)((total + TB - 1) / TB), TB, 0, stream>>>(
        H, dinv, AGG, NNODES, 8);
    long etotal = (long)NEDGES << 8;
    edge_agg_kernel<<<(int)((etotal + TB - 1) / TB), TB, 0, stream>>>(
        src, dst, dinv, H, AGG, NEDGES, 8);
    bias_relu_kernel<<<(int)((total + TB - 1) / TB), TB, 0, stream>>>(
        AGG, b1, NNODES, 8);
  }

  // --- layer 2: H2 = h1 @ W2  (reuse H buffer), aggregate into out --------
  wmma_gemm_f32_lds_kernel<DHID, DOUT><<<NNODES / 16, 128, 0, stream>>>(AGG, W2, H);
  {
    long total = (long)NNODES << 7;  // N * 128
    out_init_kernel<<<(int)((total + TB - 1) / TB), TB, 0, stream>>>(
        H, dinv, b2, x, out, NNODES, 7);
    long etotal = (long)NEDGES << 7;
    edge_agg_kernel<<<(int)((etotal + TB - 1) / TB), TB, 0, stream>>>(
        src, dst, dinv, H, out, NEDGES, 7);
  }
}